// EnhancedContextAwareDualVQ_24902220382578
// MI455X (gfx1250) — compile-verified
//
#include <hip/hip_runtime.h>
#include <hip/hip_bf16.h>

// Problem constants (from reference): B=32, L=1024, D=512, K=4096
#define NROWS 32768   // B*L
#define DIM   512
#define KCODE 4096
#define ROWS_PER_WG 64
#define CODES_PER_TILE 32   // double-buffered in LDS, 2 accumulators per wave

typedef __attribute__((ext_vector_type(2))) float v2f;
typedef __attribute__((ext_vector_type(8))) float v8f;

__device__ __forceinline__ void wait_async0() {
#if __has_builtin(__builtin_amdgcn_s_wait_asynccnt)
    __builtin_amdgcn_s_wait_asynccnt(0);
#else
    asm volatile("s_wait_asynccnt 0x0" ::: "memory");
#endif
}

// Issue async global->LDS copies for one 32-code embedding tile (64 KB).
// Each lane moves 16B per instruction; 32 instructions per wave, ASYNCcnt<=32.
__device__ __forceinline__ void async_load_tile(const float* __restrict__ gsrc,
                                                float* __restrict__ ldst, int tid) {
    #pragma unroll 8
    for (int i = tid; i < CODES_PER_TILE * DIM / 4; i += 128) {
        const int r  = i >> 7;                 // code row in tile (0..31)
        const int k4 = (i & 127) << 2;         // k (multiple of 4)
        const unsigned lds_addr =
            (unsigned)(size_t)(ldst + (r * DIM + (k4 ^ ((r & 15) << 2))));
        const unsigned goff = (unsigned)i << 4;  // byte offset from gsrc
        asm volatile("global_load_async_to_lds_b128 %0, %1, %2"
                     :: "v"(lds_addr), "v"(goff), "s"(gsrc)
                     : "memory");
    }
}

// ---------------------------------------------------------------------------
// Kernel 1: e_norms[k] = sum_d emb[k][d]^2.  One wave per code row.
// ---------------------------------------------------------------------------
__global__ __launch_bounds__(256) void enorm_kernel(const float* __restrict__ emb,
                                                    float* __restrict__ enorm) {
    const int wid  = threadIdx.x >> 5;
    const int lane = threadIdx.x & 31;
    const int row  = blockIdx.x * 8 + wid;
    const float4* e4 = (const float4*)(emb + (size_t)row * DIM);
    float s = 0.0f;
    #pragma unroll
    for (int i = lane; i < DIM / 4; i += 32) {
        float4 v = e4[i];
        s += v.x * v.x + v.y * v.y + v.z * v.z + v.w * v.w;
    }
    #pragma unroll
    for (int off = 16; off; off >>= 1) s += __shfl_xor(s, off, 32);
    if (lane == 0) enorm[row] = s;
}

// ---------------------------------------------------------------------------
// Kernel 2: argmin over codes via f32 WMMA + async double-buffered E tiles.
// ---------------------------------------------------------------------------
__global__ __launch_bounds__(128) void vq_argmin(const float* __restrict__ z,
                                                 const float* __restrict__ emb,
                                                 const float* __restrict__ enorm,
                                                 int* __restrict__ indices) {
    __shared__ float ldsX[ROWS_PER_WG * DIM];           // 128 KB, k ^= (r&15)<<2
    __shared__ float ldsE[2 * CODES_PER_TILE * DIM];    // 2 x 64 KB double buffer

    const int tid  = threadIdx.x;
    const int wave = tid >> 5;
    const int lane = tid & 31;
    const int m    = lane & 15;                // A-row / B-col index for this lane
    const int kk   = (lane >> 4) << 1;         // lanes 0-15 -> K=0,1 ; 16-31 -> K=2,3
    const int rowBase = blockIdx.x * ROWS_PER_WG;

    // Stage the 64-row X block (contiguous global float4 reads, swizzled store).
    const float4* gx4 = (const float4*)(z + (size_t)rowBase * DIM);
    #pragma unroll 8
    for (int i = tid; i < ROWS_PER_WG * DIM / 4; i += 128) {
        int r  = i >> 7;
        int k4 = (i & 127) << 2;
        float4 v = gx4[i];
        *(float4*)&ldsX[r * DIM + (k4 ^ ((r & 15) << 2))] = v;
    }

    // Prefetch embedding tile 0 into buffer 0 (async, overlaps nothing yet).
    async_load_tile(emb, &ldsE[0], tid);

    float bs[8];
    int   bi[8];
    #pragma unroll
    for (int r = 0; r < 8; ++r) { bs[r] = 3.4e38f; bi[r] = 0; }

    const int xrow = wave * 16 + m;            // all 32 lanes read rows wave*16+m
    const int xsw  = m << 2;                   // (xrow & 15) == m, same swizzle for E

    const int NTILES = KCODE / CODES_PER_TILE; // 128
    for (int ct = 0; ct < NTILES; ++ct) {
        wait_async0();        // this wave's async writes for tile ct are in LDS
        __syncthreads();      // all waves' writes visible; prev compute finished
        if (ct + 1 < NTILES)  // prefetch next tile into the other buffer
            async_load_tile(emb + (size_t)(ct + 1) * CODES_PER_TILE * DIM,
                            &ldsE[(ct & 1) ? 0 : CODES_PER_TILE * DIM], tid);

        const float* eb = &ldsE[(ct & 1) ? CODES_PER_TILE * DIM : 0];
        v8f acc0 = {}, acc1 = {};
        #pragma unroll 4
        for (int k0 = 0; k0 < DIM; k0 += 4) {
            const int kidx = (k0 + kk) ^ xsw;
            v2f a  = *(const v2f*)&ldsX[xrow * DIM + kidx];     // A: 16x4 of x
            v2f b0 = *(const v2f*)&eb[m * DIM + kidx];          // B: codes 0..15
            v2f b1 = *(const v2f*)&eb[(16 + m) * DIM + kidx];   // B: codes 16..31
            acc0 = __builtin_amdgcn_wmma_f32_16x16x4_f32(
                false, a, false, b0, (short)0, acc0, false, false);
            acc1 = __builtin_amdgcn_wmma_f32_16x16x4_f32(
                false, a, false, b1, (short)0, acc1, false, false);
        }

        const float en0 = enorm[ct * CODES_PER_TILE + m];
        const float en1 = enorm[ct * CODES_PER_TILE + 16 + m];
        #pragma unroll
        for (int r = 0; r < 8; ++r) {
            float s0 = fmaf(-2.0f, acc0[r], en0);
            if (s0 < bs[r]) { bs[r] = s0; bi[r] = ct * CODES_PER_TILE + m; }
            float s1 = fmaf(-2.0f, acc1[r], en1);
            if (s1 < bs[r]) { bs[r] = s1; bi[r] = ct * CODES_PER_TILE + 16 + m; }
        }
    }

    // Reduce (min, argmin) across the 16 lanes of each half.
    // C layout: VGPR r -> M=r (lanes 0-15), M=8+r (lanes 16-31), N=lane&15.
    #pragma unroll
    for (int off = 1; off < 16; off <<= 1) {
        #pragma unroll
        for (int r = 0; r < 8; ++r) {
            float os = __shfl_xor(bs[r], off, 32);
            int   oi = __shfl_xor(bi[r], off, 32);
            if (os < bs[r] || (os == bs[r] && oi < bi[r])) { bs[r] = os; bi[r] = oi; }
        }
    }
    if (m == 0) {
        const int mrow = rowBase + wave * 16 + ((lane >> 4) << 3);
        #pragma unroll
        for (int r = 0; r < 8; ++r) indices[mrow + r] = bi[r];
    }
}

// ---------------------------------------------------------------------------
// Kernel 3: gather quantized = emb[idx], accumulate sum((q - z)^2) per block.
// ---------------------------------------------------------------------------
__global__ __launch_bounds__(256) void vq_gather_loss(const float* __restrict__ emb,
                                                      const float* __restrict__ z,
                                                      const int* __restrict__ idx,
                                                      float* __restrict__ outq,
                                                      float* __restrict__ partial) {
    const int tid = threadIdx.x;
    const int bid = blockIdx.x;
    const size_t base = (size_t)bid * 16 * DIM;
    const float4* z4 = (const float4*)(z + base);
    float4*       o4 = (float4*)(outq + base);

    float local = 0.0f;
    #pragma unroll
    for (int i = tid; i < 16 * DIM / 4; i += 256) {
        const int row  = i >> 7;
        const int code = idx[bid * 16 + row];
        const int d4   = i & 127;
        float4 e  = ((const float4*)(emb + (size_t)code * DIM))[d4];
        float4 zz = z4[i];
        o4[i] = e;
        float dx = e.x - zz.x, dy = e.y - zz.y, dz = e.z - zz.z, dw = e.w - zz.w;
        local += dx * dx + dy * dy + dz * dz + dw * dw;
    }

    __shared__ float red[8];
    #pragma unroll
    for (int off = 16; off; off >>= 1) local += __shfl_xor(local, off, 32);
    if ((tid & 31) == 0) red[tid >> 5] = local;
    __syncthreads();
    if (tid == 0) {
        float s = 0.0f;
        #pragma unroll
        for (int w = 0; w < 8; ++w) s += red[w];
        partial[bid] = s;
    }
}

// ---------------------------------------------------------------------------
// Kernel 4: pass z_imag straight through (vectorized copy).
// ---------------------------------------------------------------------------
__global__ __launch_bounds__(256) void copy_imag(const float* __restrict__ zi,
                                                 float* __restrict__ out) {
    const float4* s4 = (const float4*)zi;
    float4*       o4 = (float4*)out;
    const int stride = gridDim.x * blockDim.x;
    for (int i = blockIdx.x * blockDim.x + threadIdx.x; i < (NROWS * DIM) / 4; i += stride)
        o4[i] = s4[i];
}

// ---------------------------------------------------------------------------
// Kernel 5: loss = 1.25 * sum(partials) / (N*D), fixed-order reduction.
// ---------------------------------------------------------------------------
__global__ __launch_bounds__(256) void final_loss(const float* __restrict__ partial,
                                                  float* __restrict__ outl) {
    __shared__ float red[256];
    float s = 0.0f;
    for (int i = threadIdx.x; i < 2048; i += 256) s += partial[i];
    red[threadIdx.x] = s;
    __syncthreads();
    for (int off = 128; off; off >>= 1) {
        if (threadIdx.x < off) red[threadIdx.x] += red[threadIdx.x + off];
        __syncthreads();
    }
    if (threadIdx.x == 0)
        outl[0] = 1.25f * red[0] / (float)((size_t)NROWS * DIM);
}

// ---------------------------------------------------------------------------
extern "C" void kernel_launch(void* const* d_in, const int* in_sizes, int n_in,
                              void* d_out, int out_size, void* d_ws, size_t ws_size,
                              hipStream_t stream) {
    const float* z_real = (const float*)d_in[0];
    const float* z_imag = (const float*)d_in[1];
    const float* emb    = (const float*)d_in[2];

    float* out  = (float*)d_out;
    float* outq = out;                              // [N*D] quantized_st
    float* outi = out + (size_t)NROWS * DIM;        // [N*D] z_imag passthrough
    float* outl = out + (size_t)2 * NROWS * DIM;    // [1]   loss

    // Workspace layout: e_norms (16KB) | indices (128KB) | partials (8KB)
    float* enorm   = (float*)d_ws;
    int*   indices = (int*)((char*)d_ws + KCODE * sizeof(float));
    float* partial = (float*)((char*)d_ws + KCODE * sizeof(float) + NROWS * sizeof(int));

    enorm_kernel  <<<KCODE / 8, 256, 0, stream>>>(emb, enorm);
    vq_argmin     <<<NROWS / ROWS_PER_WG, 128, 0, stream>>>(z_real, emb, enorm, indices);
    vq_gather_loss<<<NROWS / 16, 256, 0, stream>>>(emb, z_real, indices, outq, partial);
    copy_imag     <<<4096, 256, 0, stream>>>(z_imag, outi);
    final_loss    <<<1, 256, 0, stream>>>(partial, outl);
}